// SelectiveSSM_58059367907463
// MI455X (gfx1250) — compile-verified
//
#include <hip/hip_runtime.h>
#include <math.h>

typedef __attribute__((ext_vector_type(2))) float v2f;
typedef __attribute__((ext_vector_type(4))) float v4f;
typedef __attribute__((ext_vector_type(8))) float v8f;
typedef __attribute__((ext_vector_type(4))) int   v4i;

#ifndef __has_builtin
#define __has_builtin(x) 0
#endif
#if __has_builtin(__builtin_amdgcn_global_load_async_to_lds_b128)
#define HAS_ASYNC 1
#else
#define HAS_ASYNC 0
#endif

/* Async DMA global->LDS (ASYNCcnt path). 16B-aligned src and dst required.
 * Builtin signature (per hipcc diagnostic): param0 is generic `int4*` (the
 * global source, following the __builtin_amdgcn_global_load_lds precedent of
 * (src, dst, imm offset, imm cpol)). */
__device__ __forceinline__ void async_b128(const float* g, float* l) {
#if HAS_ASYNC
  __builtin_amdgcn_global_load_async_to_lds_b128((v4i*)g, (v4i*)l, 0, 0);
#else
  *(v4f*)l = *(const v4f*)g; /* synchronous fallback */
#endif
}
__device__ __forceinline__ void wait_async_all() {
#if HAS_ASYNC
#if __has_builtin(__builtin_amdgcn_s_wait_asynccnt)
  __builtin_amdgcn_s_wait_asynccnt(0);
#else
  asm volatile("s_wait_asynccnt 0" ::: "memory");
#endif
#endif
}

#define D_MODEL 768
#define STATE   64
#define D_INNER 1536
#define BATCH   2
#define SEQ     1024
#define ROWS    (BATCH*SEQ)   /* 2048 */

/* ---------------- fp32 WMMA GEMM: C = A(MxK,lda) @ B(KxN,ldb) ----------------
 * 256 threads (8 waves, wave32). Block tile 128x64, BK=16, double-buffered LDS
 * filled by GLOBAL_LOAD_ASYNC_TO_LDS_B128; one barrier per K-tile.
 * Waves 4x2, each computes 32x32 = 2x2 tiles of V_WMMA_F32_16X16X4_F32.
 * EPI==1: C = softplus(A@B + bias[col]). */
#define BM 128
#define BN 64
#define BK 16
#define LDA_S 36   /* 144B row stride: 16B-aligned b128 stores, conflict-free b64 frags */
#define LDB_S 68   /* 272B row stride: 16B-aligned */

template<int EPI>
__global__ __launch_bounds__(256)
void gemm_f32_wmma(const float* __restrict__ A, const float* __restrict__ B,
                   const float* __restrict__ bias, float* __restrict__ C,
                   int M, int N, int K, int lda, int ldb, int ldc)
{
  __shared__ float As[2][BM * LDA_S];
  __shared__ float Bs[2][BK * LDB_S];

  const int tid  = threadIdx.x;
  const int lane = tid & 31;
  const int wave = tid >> 5;
  const int wm   = wave >> 1;        /* 0..3 */
  const int wn   = wave & 1;         /* 0..1 */
  const int m0   = blockIdx.y * BM;
  const int n0   = blockIdx.x * BN;

  const int fm  = lane & 15;
  const int fkh = lane >> 4;

  /* per-thread staging coordinates (3 async b128 per thread per tile) */
  const int ar0 = tid >> 2,         acv = (tid & 3) << 2;       /* A: idx=tid     */
  const int ar1 = (tid + 256) >> 2;                              /* A: idx=tid+256 */
  const int br  = tid >> 4,         bcv = (tid & 15) << 2;       /* B tile         */

  v8f c[2][2] = {};

  /* prologue: DMA tile 0 into buffer 0 */
  async_b128(A + (size_t)(m0 + ar0) * lda + acv,        &As[0][ar0 * LDA_S + acv]);
  async_b128(A + (size_t)(m0 + ar1) * lda + acv,        &As[0][ar1 * LDA_S + acv]);
  async_b128(B + (size_t)br * ldb + n0 + bcv,           &Bs[0][br * LDB_S + bcv]);

  for (int k0 = 0; k0 < K; k0 += BK) {
    const int cur = (k0 / BK) & 1;
    wait_async_all();      /* this wave's DMA for tile k complete          */
    __syncthreads();       /* all waves: DMA landed + prev compute retired */

    if (k0 + BK < K) {     /* kick off DMA for tile k+1 into other buffer */
      const int nxt = cur ^ 1;
      const int kn  = k0 + BK;
      async_b128(A + (size_t)(m0 + ar0) * lda + kn + acv, &As[nxt][ar0 * LDA_S + acv]);
      async_b128(A + (size_t)(m0 + ar1) * lda + kn + acv, &As[nxt][ar1 * LDA_S + acv]);
      async_b128(B + (size_t)(kn + br) * ldb + n0 + bcv,  &Bs[nxt][br * LDB_S + bcv]);
    }

#pragma unroll
    for (int kk = 0; kk < BK; kk += 4) {
      v2f a[2];
#pragma unroll
      for (int i = 0; i < 2; ++i)
        a[i] = *(const v2f*)(&As[cur][(wm * 32 + i * 16 + fm) * LDA_S + kk + fkh * 2]);
      v2f b[2];
#pragma unroll
      for (int j = 0; j < 2; ++j) {
        int cb = wn * 32 + j * 16 + fm;
        b[j].x = Bs[cur][(kk + fkh * 2    ) * LDB_S + cb];
        b[j].y = Bs[cur][(kk + fkh * 2 + 1) * LDB_S + cb];
      }
#pragma unroll
      for (int i = 0; i < 2; ++i)
#pragma unroll
        for (int j = 0; j < 2; ++j)
          c[i][j] = __builtin_amdgcn_wmma_f32_16x16x4_f32(
              false, a[i], false, b[j], (short)0, c[i][j], false, false);
    }
  }

  /* C/D layout: vgpr v, lanes 0-15 -> M=v, lanes 16-31 -> M=v+8, N=lane&15 */
#pragma unroll
  for (int i = 0; i < 2; ++i) {
#pragma unroll
    for (int j = 0; j < 2; ++j) {
      const int col = n0 + wn * 32 + j * 16 + fm;
#pragma unroll
      for (int v = 0; v < 8; ++v) {
        const int row = m0 + wm * 32 + i * 16 + fkh * 8 + v;
        float val = c[i][j][v];
        if (EPI == 1) {
          float xv = val + bias[col];
          val = fmaxf(xv, 0.0f) + log1pf(__expf(-fabsf(xv)));
        }
        C[(size_t)row * ldc + col] = val;
      }
    }
  }
}

/* ---------------- selective scan -----------------------------------------
 * One lane per (batch, channel). h[64] in VGPRs. B_t/C_t staged via async DMA
 * into double-buffered LDS chunks of 16 timesteps; one barrier per chunk.
 * Fuses y = (scan + x*D) * silu(z). */
#define TCHUNK 16

__global__ __launch_bounds__(256)
void scan_kernel(const float* __restrict__ xz, const float* __restrict__ dtb,
                 const float* __restrict__ bc, const float* __restrict__ A_log,
                 const float* __restrict__ Dv, float* __restrict__ y)
{
  __shared__ float sA[STATE];
  __shared__ float sBC[2][TCHUNK][2 * STATE];

  const int d = blockIdx.x * 256 + threadIdx.x;  /* channel 0..1535 */
  const int b = blockIdx.y;

  if (threadIdx.x < STATE) sA[threadIdx.x] = -__expf(A_log[threadIdx.x]);

  /* staging coords: chunk = TCHUNK*128 floats = 512 b128, 2 per thread */
  const int t_tt0 = threadIdx.x >> 5,         t_jv = (threadIdx.x & 31) << 2;
  const int t_tt1 = (threadIdx.x + 256) >> 5;
  const float* bcb = bc + (size_t)b * SEQ * (2 * STATE);

  /* prologue: DMA chunk 0 */
  async_b128(bcb + (size_t)t_tt0 * (2 * STATE) + t_jv, &sBC[0][t_tt0][t_jv]);
  async_b128(bcb + (size_t)t_tt1 * (2 * STATE) + t_jv, &sBC[0][t_tt1][t_jv]);

  float h[STATE];
#pragma unroll
  for (int n = 0; n < STATE; ++n) h[n] = 0.0f;
  const float Dd = Dv[d];

  for (int t0 = 0; t0 < SEQ; t0 += TCHUNK) {
    const int cur = (t0 / TCHUNK) & 1;
    wait_async_all();
    __syncthreads();

    if (t0 + TCHUNK < SEQ) {
      const int nxt = cur ^ 1;
      const float* src = bcb + (size_t)(t0 + TCHUNK) * (2 * STATE);
      async_b128(src + (size_t)t_tt0 * (2 * STATE) + t_jv, &sBC[nxt][t_tt0][t_jv]);
      async_b128(src + (size_t)t_tt1 * (2 * STATE) + t_jv, &sBC[nxt][t_tt1][t_jv]);
    }

    for (int tt = 0; tt < TCHUNK; ++tt) {
      const int r = b * SEQ + t0 + tt;
      const float xv  = xz[(size_t)r * (2 * D_INNER) + d];
      const float zv  = xz[(size_t)r * (2 * D_INNER) + D_INNER + d];
      const float dt  = dtb[(size_t)r * D_INNER + d];
      const float dtx = dt * xv;
      float acc = 0.0f;
#pragma unroll
      for (int n = 0; n < STATE; ++n) {
        float dA = __expf(dt * sA[n]);            /* broadcast LDS reads */
        h[n] = fmaf(dA, h[n], dtx * sBC[cur][tt][n]);
        acc  = fmaf(h[n], sBC[cur][tt][STATE + n], acc);
      }
      const float yv  = acc + xv * Dd;
      const float sig = 1.0f / (1.0f + __expf(-zv));
      y[(size_t)r * D_INNER + d] = yv * (zv * sig);
    }
  }
}

/* ---------------- driver ------------------------------------------------- */
extern "C" void kernel_launch(void* const* d_in, const int* in_sizes, int n_in,
                              void* d_out, int out_size, void* d_ws, size_t ws_size,
                              hipStream_t stream)
{
  const float* x     = (const float*)d_in[0];
  const float* W_in  = (const float*)d_in[1];
  const float* W_x   = (const float*)d_in[2];
  const float* W_dt  = (const float*)d_in[3];
  const float* b_dt  = (const float*)d_in[4];
  const float* A_log = (const float*)d_in[5];
  const float* Dv    = (const float*)d_in[6];
  const float* W_out = (const float*)d_in[7];
  float* out = (float*)d_out;

  /* workspace layout (floats):
   * xz : 2048*3072  (xp = cols [0,1536), z = cols [1536,3072))
   * dt : 2048*1536
   * bc : 2048*128   (B = cols [0,64), C = cols [64,128))
   * y  : 2048*1536   total ~51.4 MB */
  float* xzb = (float*)d_ws;
  float* dtb = xzb + (size_t)ROWS * 2 * D_INNER;
  float* bcb = dtb + (size_t)ROWS * D_INNER;
  float* yb  = bcb + (size_t)ROWS * 2 * STATE;

  dim3 blk(256);

  /* 1. xz = x @ W_in : 2048x768 @ 768x3072 */
  gemm_f32_wmma<0><<<dim3((2 * D_INNER) / BN, ROWS / BM), blk, 0, stream>>>(
      x, W_in, nullptr, xzb, ROWS, 2 * D_INNER, D_MODEL,
      D_MODEL, 2 * D_INNER, 2 * D_INNER);

  /* 2. [B|C] = xp @ W_x : 2048x1536 @ 1536x128 */
  gemm_f32_wmma<0><<<dim3((2 * STATE) / BN, ROWS / BM), blk, 0, stream>>>(
      xzb, W_x, nullptr, bcb, ROWS, 2 * STATE, D_INNER,
      2 * D_INNER, 2 * STATE, 2 * STATE);

  /* 3. dt = softplus(xp @ W_dt + b_dt) : 2048x1536 @ 1536x1536 */
  gemm_f32_wmma<1><<<dim3(D_INNER / BN, ROWS / BM), blk, 0, stream>>>(
      xzb, W_dt, b_dt, dtb, ROWS, D_INNER, D_INNER,
      2 * D_INNER, D_INNER, D_INNER);

  /* 4. sequential selective scan + fused D-residual and silu(z) gating */
  scan_kernel<<<dim3(D_INNER / 256, BATCH), blk, 0, stream>>>(
      xzb, dtb, bcb, A_log, Dv, yb);

  /* 5. out = y @ W_out : 2048x1536 @ 1536x768 */
  gemm_f32_wmma<0><<<dim3(D_MODEL / BN, ROWS / BM), blk, 0, stream>>>(
      yb, W_out, nullptr, out, ROWS, D_MODEL, D_INNER,
      D_INNER, D_MODEL, D_MODEL);
}